// HyperpriorDensity_4982162063467
// MI455X (gfx1250) — compile-verified
//
#include <hip/hip_runtime.h>

// HyperpriorDensity forward for MI455X (gfx1250).
//
// Per-channel 1->3->3->3->1 MLP chain evaluated at x±0.5, sigmoid difference,
// clamp at 1e-9. Streaming elementwise kernel: memory floor 40MiB/23.3TB/s
// ≈ 1.8us; ~24 transcendentals/element make gfx1250 v_tanh_f32 the critical
// instruction. WMMA intentionally NOT used: 3x3 per-channel matmuls with a
// tanh between every layer have <2% tile utilization — register FMAs win.

#define CHN      320
#define WSTRIDE  48    // floats per channel in the weight workspace (44 used)

typedef float v4f __attribute__((ext_vector_type(4)));

#define LOG2E  1.4426950408889634f
#define LN2    0.6931471805599453f

__device__ __forceinline__ float ftanh(float x) {
#if __has_builtin(__builtin_amdgcn_tanhf)
  return __builtin_amdgcn_tanhf(x);            // v_tanh_f32 (gfx1250 TRANS)
#else
  // tanh(x) = 1 - 2/(e^{2x}+1), via v_exp_f32 (2^x) + v_rcp_f32
  float e = __builtin_amdgcn_exp2f(x * (2.0f * LOG2E));
  return 1.0f - 2.0f * __builtin_amdgcn_rcpf(e + 1.0f);
#endif
}

__device__ __forceinline__ float fsigmoid(float z) {
  // 1/(1+e^{-z}) via v_exp_f32 + v_rcp_f32
  return __builtin_amdgcn_rcpf(1.0f + __builtin_amdgcn_exp2f(z * -LOG2E));
}

__device__ __forceinline__ float fsoftplus(float x) {
  // log(1+e^x); weights are small so the naive form is safe, but guard anyway
  float e  = __builtin_amdgcn_exp2f(x * LOG2E);
  float sp = __builtin_amdgcn_logf(1.0f + e) * LN2;
  return (x > 20.0f) ? x : sp;
}

// ---------------------------------------------------------------------------
// Pre-pass: fold softplus(H) and tanh(a) once per channel into workspace.
// Layout per channel (stride 48 floats):
//   [ 0.. 2] sp(H0)   [ 3..11] sp(H1)  [12..20] sp(H2)  [21..23] sp(H3)
//   [24..26] th(a0)   [27..29] th(a1)  [30..32] th(a2)  [33]     th(a3)
//   [34..36] b0       [37..39] b1      [40..42] b2      [43]     b3
// ---------------------------------------------------------------------------
__global__ void prep_weights(const float* __restrict__ H0, const float* __restrict__ a0, const float* __restrict__ b0,
                             const float* __restrict__ H1, const float* __restrict__ a1, const float* __restrict__ b1,
                             const float* __restrict__ H2, const float* __restrict__ a2, const float* __restrict__ b2,
                             const float* __restrict__ H3, const float* __restrict__ a3, const float* __restrict__ b3,
                             float* __restrict__ w) {
  int c = blockIdx.x * blockDim.x + threadIdx.x;
  if (c >= CHN) return;
  float* o = w + c * WSTRIDE;
#pragma unroll
  for (int j = 0; j < 3; ++j) o[0 + j]  = fsoftplus(H0[c * 3 + j]);
#pragma unroll
  for (int j = 0; j < 9; ++j) o[3 + j]  = fsoftplus(H1[c * 9 + j]);
#pragma unroll
  for (int j = 0; j < 9; ++j) o[12 + j] = fsoftplus(H2[c * 9 + j]);
#pragma unroll
  for (int j = 0; j < 3; ++j) o[21 + j] = fsoftplus(H3[c * 3 + j]);
#pragma unroll
  for (int j = 0; j < 3; ++j) o[24 + j] = ftanh(a0[c * 3 + j]);
#pragma unroll
  for (int j = 0; j < 3; ++j) o[27 + j] = ftanh(a1[c * 3 + j]);
#pragma unroll
  for (int j = 0; j < 3; ++j) o[30 + j] = ftanh(a2[c * 3 + j]);
  o[33] = ftanh(a3[c]);
#pragma unroll
  for (int j = 0; j < 3; ++j) o[34 + j] = b0[c * 3 + j];
#pragma unroll
  for (int j = 0; j < 3; ++j) o[37 + j] = b1[c * 3 + j];
#pragma unroll
  for (int j = 0; j < 3; ++j) o[40 + j] = b2[c * 3 + j];
  o[43] = b3[c];
}

// ---------------------------------------------------------------------------
// Main kernel
// ---------------------------------------------------------------------------
struct Wt {
  float h0[3], h1[9], h2[9], h3[3];
  float t0[3], t1[3], t2[3], t3;
  float b0[3], b1[3], b2[3], b3;
};

__device__ __forceinline__ float chain(float v, const Wt& W) {
  // layer 0: 1 -> 3
  float p0 = fmaf(W.h0[0], v, W.b0[0]);
  float p1 = fmaf(W.h0[1], v, W.b0[1]);
  float p2 = fmaf(W.h0[2], v, W.b0[2]);
  p0 = fmaf(W.t0[0], ftanh(p0), p0);
  p1 = fmaf(W.t0[1], ftanh(p1), p1);
  p2 = fmaf(W.t0[2], ftanh(p2), p2);
  // layer 1: 3 -> 3
  float q0 = fmaf(W.h1[0], p0, fmaf(W.h1[1], p1, fmaf(W.h1[2], p2, W.b1[0])));
  float q1 = fmaf(W.h1[3], p0, fmaf(W.h1[4], p1, fmaf(W.h1[5], p2, W.b1[1])));
  float q2 = fmaf(W.h1[6], p0, fmaf(W.h1[7], p1, fmaf(W.h1[8], p2, W.b1[2])));
  q0 = fmaf(W.t1[0], ftanh(q0), q0);
  q1 = fmaf(W.t1[1], ftanh(q1), q1);
  q2 = fmaf(W.t1[2], ftanh(q2), q2);
  // layer 2: 3 -> 3
  float r0 = fmaf(W.h2[0], q0, fmaf(W.h2[1], q1, fmaf(W.h2[2], q2, W.b2[0])));
  float r1 = fmaf(W.h2[3], q0, fmaf(W.h2[4], q1, fmaf(W.h2[5], q2, W.b2[1])));
  float r2 = fmaf(W.h2[6], q0, fmaf(W.h2[7], q1, fmaf(W.h2[8], q2, W.b2[2])));
  r0 = fmaf(W.t2[0], ftanh(r0), r0);
  r1 = fmaf(W.t2[1], ftanh(r1), r1);
  r2 = fmaf(W.t2[2], ftanh(r2), r2);
  // layer 3: 3 -> 1
  float o = fmaf(W.h3[0], r0, fmaf(W.h3[1], r1, fmaf(W.h3[2], r2, W.b3)));
  o = fmaf(W.t3, ftanh(o), o);
  return o;
}

__device__ __forceinline__ float density(float v, const Wt& W) {
  float u = chain(v + 0.5f, W);   // cdf_upper logits
  float l = chain(v - 0.5f, W);   // cdf_lower logits
  float sum = u + l;
  float s = (sum > 0.0f) ? -1.0f : ((sum < 0.0f) ? 1.0f : 0.0f);  // -sign(sum)
  float like = fabsf(fsigmoid(s * u) - fsigmoid(s * l));
  return fmaxf(like, 1e-9f);
}

// One block per 1024-element channel plane: blockIdx.x = n*C + c, so the
// channel (and the 44 weight loads) are block-uniform -> SGPR s_loads.
// 256 threads x float4 NT loads/stores stream x and out exactly once.
__global__ void __launch_bounds__(256)
density_kernel(const float* __restrict__ x, const float* __restrict__ w,
               float* __restrict__ out) {
  const int plane = blockIdx.x;          // n*C + c
  const int c = plane % CHN;
  const float* __restrict__ W = w + c * WSTRIDE;

  Wt wt;
#pragma unroll
  for (int j = 0; j < 3; ++j) wt.h0[j] = W[0 + j];
#pragma unroll
  for (int j = 0; j < 9; ++j) wt.h1[j] = W[3 + j];
#pragma unroll
  for (int j = 0; j < 9; ++j) wt.h2[j] = W[12 + j];
#pragma unroll
  for (int j = 0; j < 3; ++j) wt.h3[j] = W[21 + j];
#pragma unroll
  for (int j = 0; j < 3; ++j) wt.t0[j] = W[24 + j];
#pragma unroll
  for (int j = 0; j < 3; ++j) wt.t1[j] = W[27 + j];
#pragma unroll
  for (int j = 0; j < 3; ++j) wt.t2[j] = W[30 + j];
  wt.t3 = W[33];
#pragma unroll
  for (int j = 0; j < 3; ++j) wt.b0[j] = W[34 + j];
#pragma unroll
  for (int j = 0; j < 3; ++j) wt.b1[j] = W[37 + j];
#pragma unroll
  for (int j = 0; j < 3; ++j) wt.b2[j] = W[40 + j];
  wt.b3 = W[43];

  const long base = (long)plane * 1024 + (long)threadIdx.x * 4;
  v4f xv = __builtin_nontemporal_load((const v4f*)(x + base));  // TH=NT b128

  v4f r;
  r.x = density(xv.x, wt);
  r.y = density(xv.y, wt);
  r.z = density(xv.z, wt);
  r.w = density(xv.w, wt);

  __builtin_nontemporal_store(r, (v4f*)(out + base));           // TH=NT b128
}

// ---------------------------------------------------------------------------
// Entry point
// ---------------------------------------------------------------------------
extern "C" void kernel_launch(void* const* d_in, const int* in_sizes, int n_in,
                              void* d_out, int out_size, void* d_ws, size_t ws_size,
                              hipStream_t stream) {
  (void)in_sizes; (void)n_in; (void)ws_size;
  // setup_inputs() dict order: x, H0,a0,b0, H1,a1,b1, H2,a2,b2, H3,a3,b3
  const float* x  = (const float*)d_in[0];
  const float* H0 = (const float*)d_in[1];
  const float* a0 = (const float*)d_in[2];
  const float* b0 = (const float*)d_in[3];
  const float* H1 = (const float*)d_in[4];
  const float* a1 = (const float*)d_in[5];
  const float* b1 = (const float*)d_in[6];
  const float* H2 = (const float*)d_in[7];
  const float* a2 = (const float*)d_in[8];
  const float* b2 = (const float*)d_in[9];
  const float* H3 = (const float*)d_in[10];
  const float* a3 = (const float*)d_in[11];
  const float* b3 = (const float*)d_in[12];
  float* out = (float*)d_out;
  float* w   = (float*)d_ws;   // needs CHN*WSTRIDE*4 = 60 KiB of workspace

  prep_weights<<<(CHN + 127) / 128, 128, 0, stream>>>(
      H0, a0, b0, H1, a1, b1, H2, a2, b2, H3, a3, b3, w);

  const int planes = out_size / 1024;    // N*C = 5120 channel planes
  density_kernel<<<planes, 256, 0, stream>>>(x, w, out);
}